// CCM_22213570855326
// MI455X (gfx1250) — compile-verified
//
#include <hip/hip_runtime.h>

// Problem constants (match setup_inputs): m:(8,27,1000,257) f32, x:(8,257,1000,2) f32,
// v:(2,3) f32. out:(8,257,1000,2) f32.
#define BATCH 8
#define CHANS 27
#define TDIM 1000
#define FDIM 257

#define BT 16              // time tile
#define BF 32              // freq tile
#define XT (BT + 2)        // x tile time extent (halo: t-2..t)
#define XF (BF + 2)        // x tile freq extent (halo: f-1..f+1)
#define XROW (XT * 2)      // floats per LDS x-tile row  (= 36 floats = 144 bytes)
#define OROW (BF * 2 + 2)  // padded output-staging row (floats)

typedef int vi2 __attribute__((vector_size(8)));        // matches async builtin param type
typedef float fv2 __attribute__((ext_vector_type(2)));  // native vec for NT store
typedef unsigned int u32x4 __attribute__((vector_size(16)));
typedef int i32x4 __attribute__((vector_size(16)));
typedef int i32x8 __attribute__((vector_size(32)));

#if defined(__HIP_DEVICE_COMPILE__) && __has_builtin(__builtin_amdgcn_global_load_async_to_lds_b64)
#define HAVE_ASYNC_LDS 1
#else
#define HAVE_ASYNC_LDS 0
#endif

#if defined(__HIP_DEVICE_COMPILE__) && __has_builtin(__builtin_amdgcn_tensor_load_to_lds)
#define HAVE_TDM 1
#else
#define HAVE_TDM 0
#endif

// Copy 8 bytes global -> LDS. Prefers the CDNA5 async-to-LDS path (ASYNCcnt).
__device__ __forceinline__ void xcopy8(const float* __restrict__ src, float* lds_dst) {
#if HAVE_ASYNC_LDS
  __builtin_amdgcn_global_load_async_to_lds_b64(
      (__attribute__((address_space(1))) vi2*)(size_t)src,
      (__attribute__((address_space(3))) vi2*)lds_dst,
      /*imm offset*/ 0, /*cpol*/ 0);
#else
  lds_dst[0] = src[0];
  lds_dst[1] = src[1];
#endif
}

__device__ __forceinline__ void wait_async_lds() {
#if HAVE_ASYNC_LDS
#if __has_builtin(__builtin_amdgcn_s_wait_asynccnt)
  __builtin_amdgcn_s_wait_asynccnt(0);
#else
  asm volatile("s_wait_asynccnt 0" ::: "memory");
#endif
#endif
}

// Boundary-block staging: per-element async copies, zeros for out-of-range (reference padding).
// LDS layout: xtile[fi][ti*2+c]  (f outer, (t,c) inner -> matches TDM row packing).
__device__ __forceinline__ void stage_x_tile_async(const float* __restrict__ x,
                                                   float (*xtile)[XROW],
                                                   int b, int t0, int f0, int tid) {
  for (int e = tid; e < XT * XF; e += BF * BT) {
    const int fi = e / XT;
    const int ti = e - fi * XT;  // ti inner: global addresses contiguous along (t,c)
    const int tg = t0 - 2 + ti;
    const int fg = f0 - 1 + fi;
    float* dst = &xtile[fi][ti * 2];
    if ((unsigned)tg < (unsigned)TDIM && (unsigned)fg < (unsigned)FDIM) {
      const float* src = x + (((size_t)b * FDIM + fg) * TDIM + tg) * 2;
      xcopy8(src, dst);  // async DMA into LDS (disjoint from the zeroed entries)
    } else {
      dst[0] = 0.0f;
      dst[1] = 0.0f;
    }
  }
  wait_async_lds();
}

__global__ __launch_bounds__(BF * BT) void ccm_deepfilter_kernel(
    const float* __restrict__ m, const float* __restrict__ x,
    const float* __restrict__ v, float* __restrict__ out) {
  __shared__ float xtile[XF][XROW];  // [f][t,c] : TDM rows pack contiguously (144 B each)
  __shared__ float otile[BT][OROW];  // padded rows for transposed store

  const int tx = threadIdx.x;    // 0..BF-1 (freq)
  const int ty = threadIdx.y;    // 0..BT-1 (time)
  const int tid = ty * BF + tx;  // 0..511
  const int f0 = blockIdx.x * BF;
  const int t0 = blockIdx.y * BT;
  const int b = blockIdx.z;

  // Interior blocks: whole halo window in-bounds -> single TDM 2D-tile DMA by wave 0.
  const bool interior =
      (t0 >= 2) && (t0 + BT <= TDIM) && (f0 >= 1) && (f0 + BF + 1 <= FDIM);

#if HAVE_TDM
  if (interior) {
    if (tid < 32) {  // wave 0 issues the tensor DMA (EXEC-independent, SGPR descriptor)
      const unsigned lds_base =
          (unsigned)(size_t)(__attribute__((address_space(3))) void*)&xtile[0][0];
      const float* gsrc = x + (((size_t)b * FDIM + (f0 - 1)) * TDIM + (t0 - 2)) * 2;
      const unsigned long long ga = (unsigned long long)(size_t)gsrc;

      // D# group 0: count=1 | lds_addr | global_addr[56:0] | type=2
      u32x4 g0 = {1u, lds_base, (unsigned)(ga & 0xFFFFFFFFu),
                  (unsigned)((ga >> 32) & 0x01FFFFFFu) | (2u << 30)};
      // D# group 1 (bit-packed per ISA 8.4), data_size=3 (8 bytes = one float2):
      //   tensor_dim0 = tile_dim0 = XT elements, tensor_dim1 = tile_dim1 = XF rows
      //   tensor_dim0_stride = TDIM elements (row-to-row = T*8 bytes); no pad/iterate/mask
      i32x8 g1 = {(int)(3u << 16),               // [17:16] data_size=8B, mask=0
                  (int)((unsigned)XT << 16),     // [79:48] tensor_dim0 lo16
                  (int)((unsigned)XF << 16),     // tensor_dim0 hi16=0 | [111:80] tensor_dim1 lo16
                  (int)((unsigned)XT << 16),     // tensor_dim1 hi16=0 | [127:112] tile_dim0
                  (int)XF,                       // [143:128] tile_dim1, tile_dim2=0
                  (int)TDIM,                     // [207:160] tensor_dim0_stride lo32
                  0,                             // stride hi16 | tensor_dim1_stride lo16 (unused)
                  0};                            // tensor_dim1_stride hi32 (unused)
      i32x4 gz4 = {0, 0, 0, 0};                  // groups 2/3: 2D tensor -> zeros
      i32x8 gz8 = {0, 0, 0, 0, 0, 0, 0, 0};      // trailing group (6-arg form) -> zeros
      __builtin_amdgcn_tensor_load_to_lds(g0, g1, gz4, gz4, gz8, /*cpol*/ 0);
#if __has_builtin(__builtin_amdgcn_s_wait_tensorcnt)
      __builtin_amdgcn_s_wait_tensorcnt(0);
#else
      asm volatile("s_wait_tensorcnt 0" ::: "memory");
#endif
    }
  } else {
    stage_x_tile_async(x, xtile, b, t0, f0, tid);
  }
#else
  (void)interior;
  stage_x_tile_async(x, xtile, b, t0, f0, tid);
#endif
  __syncthreads();

  // ---- Compute: 9 complex taps; m stream is the dominant (non-reused) traffic -> NT loads.
  const size_t TF = (size_t)TDIM * FDIM;
  int tcl = t0 + ty;  if (tcl >= TDIM) tcl = TDIM - 1;  // clamp (masked at store)
  int fcl = f0 + tx;  if (fcl >= FDIM) fcl = FDIM - 1;
  const float* mb = m + (size_t)b * CHANS * TF + (size_t)tcl * FDIM + fcl;

  const float v00 = v[0], v01 = v[1], v02 = v[2];
  const float v10 = v[3], v11 = v[4], v12 = v[5];

  float accr = 0.0f, acci = 0.0f;
#pragma unroll
  for (int n = 0; n < 9; ++n) {
    const float a0 = __builtin_nontemporal_load(mb + (size_t)(0 + n) * TF);
    const float a1 = __builtin_nontemporal_load(mb + (size_t)(9 + n) * TF);
    const float a2 = __builtin_nontemporal_load(mb + (size_t)(18 + n) * TF);
    const float hr = fmaf(v00, a0, fmaf(v01, a1, v02 * a2));
    const float hi = fmaf(v10, a0, fmaf(v11, a1, v12 * a2));
    const int i = n / 3, j = n % 3;
    const float xr = xtile[tx + j][(ty + i) * 2 + 0];
    const float xi = xtile[tx + j][(ty + i) * 2 + 1];
    accr = fmaf(hr, xr, fmaf(-hi, xi, accr));
    acci = fmaf(hr, xi, fmaf(hi, xr, acci));
  }

  otile[ty][tx * 2 + 0] = accr;
  otile[ty][tx * 2 + 1] = acci;
  __syncthreads();

  // ---- Transposed store: lanes run along t so (t,c) is contiguous in the (B,F,T,2) output.
  const int fl = tid >> 4;  // 0..31 : freq within tile
  const int tq = tid & 15;  // 0..15 : time within tile
  const int fo = f0 + fl;
  const int to = t0 + tq;
  if (fo < FDIM && to < TDIM) {
    fv2 val;
    val.x = otile[tq][fl * 2 + 0];
    val.y = otile[tq][fl * 2 + 1];
    fv2* dst = (fv2*)(out + (((size_t)b * FDIM + fo) * TDIM + to) * 2);
    __builtin_nontemporal_store(val, dst);
  }
}

extern "C" void kernel_launch(void* const* d_in, const int* in_sizes, int n_in,
                              void* d_out, int out_size, void* d_ws, size_t ws_size,
                              hipStream_t stream) {
  (void)in_sizes; (void)n_in; (void)out_size; (void)d_ws; (void)ws_size;
  const float* m = (const float*)d_in[0];
  const float* x = (const float*)d_in[1];
  const float* v = (const float*)d_in[2];
  float* out = (float*)d_out;

  dim3 block(BF, BT, 1);                                         // 512 threads = 16 waves
  dim3 grid((FDIM + BF - 1) / BF, (TDIM + BT - 1) / BT, BATCH);  // 9 x 63 x 8
  hipLaunchKernelGGL(ccm_deepfilter_kernel, grid, block, 0, stream, m, x, v, out);
}